// GCNNet_89678917140834
// MI455X (gfx1250) — compile-verified
//
#include <hip/hip_runtime.h>
#include <hip/hip_bf16.h>

typedef float v2f __attribute__((ext_vector_type(2)));
typedef float v8f __attribute__((ext_vector_type(8)));

#define DG 128     // gcn dim
#define TILE_M 32  // nodes per GEMM block

// ---------------------------------------------------------------------------
// Stage 1: per-edge degree count + layer-0 aggregation (3 floats per edge)
// ---------------------------------------------------------------------------
__global__ void deg_agg0_kernel(const int* __restrict__ src, const int* __restrict__ dst,
                                const float* __restrict__ x,
                                float* __restrict__ deg, float* __restrict__ agg0, int E) {
  int e = blockIdx.x * blockDim.x + threadIdx.x;
  if (e >= E) return;
  int s = src[e], d = dst[e];
  atomicAdd(&deg[d], 1.0f);
  atomicAdd(&agg0[d * 3 + 0], x[s * 3 + 0]);
  atomicAdd(&agg0[d * 3 + 1], x[s * 3 + 1]);
  atomicAdd(&agg0[d * 3 + 2], x[s * 3 + 2]);
}

// ---------------------------------------------------------------------------
// Stage 2: layer-0 node update (K=3 GEMM, scalar) + ReLU + fused LayerNorm.
// One 128-thread block per node; LDS tree reduction for mean/var.
// ---------------------------------------------------------------------------
__global__ void __launch_bounds__(DG) layer0_node_kernel(
    const float* __restrict__ x, const float* __restrict__ agg0,
    const float* __restrict__ deg,
    const float* __restrict__ Wl0, const float* __restrict__ Wr0,
    const float* __restrict__ b0,
    const float* __restrict__ ln_g, const float* __restrict__ ln_b,
    float* __restrict__ h, int N) {
  int node = blockIdx.x;
  if (node >= N) return;
  int j = threadIdx.x;                  // output channel 0..127
  __shared__ float red[DG];

  float invd = 1.0f / fmaxf(deg[node], 1.0f);
  float a0 = agg0[node * 3 + 0] * invd;
  float a1 = agg0[node * 3 + 1] * invd;
  float a2 = agg0[node * 3 + 2] * invd;
  float x0 = x[node * 3 + 0];
  float x1 = x[node * 3 + 1];
  float x2 = x[node * 3 + 2];

  float t = b0[j];
  t = fmaf(a0, Wl0[0 * DG + j], t);
  t = fmaf(a1, Wl0[1 * DG + j], t);
  t = fmaf(a2, Wl0[2 * DG + j], t);
  t = fmaf(x0, Wr0[0 * DG + j], t);
  t = fmaf(x1, Wr0[1 * DG + j], t);
  t = fmaf(x2, Wr0[2 * DG + j], t);
  t = fmaxf(t, 0.0f);                   // ReLU

  // LayerNorm over the 128 channels
  red[j] = t;
  __syncthreads();
  for (int s = DG / 2; s > 0; s >>= 1) {
    if (j < s) red[j] += red[j + s];
    __syncthreads();
  }
  float mu = red[0] * (1.0f / DG);
  __syncthreads();
  float dv = t - mu;
  red[j] = dv * dv;
  __syncthreads();
  for (int s = DG / 2; s > 0; s >>= 1) {
    if (j < s) red[j] += red[j + s];
    __syncthreads();
  }
  float var = red[0] * (1.0f / DG);
  float y = dv * rsqrtf(var + 1e-5f) * ln_g[j] + ln_b[j];
  h[(size_t)node * DG + j] = y;
}

// ---------------------------------------------------------------------------
// Stage 3: layer-1 aggregation. One wave per edge; float4 coalesced gather of
// h[src], 4 no-return f32 global atomics per lane into agg1[dst].
// This is the roofline-dominant stage (~1.6 GB irregular, mostly L2-resident).
// ---------------------------------------------------------------------------
__global__ void layer1_agg_kernel(const int* __restrict__ src, const int* __restrict__ dst,
                                  const float* __restrict__ h,
                                  float* __restrict__ agg1, int E) {
  long long t = (long long)blockIdx.x * blockDim.x + threadIdx.x;
  int e = (int)(t >> 5);
  int lane = (int)(t & 31);
  if (e >= E) return;
  int s = src[e], d = dst[e];
  const float4 v = *(const float4*)(h + (size_t)s * DG + lane * 4);
  float* ap = agg1 + (size_t)d * DG + lane * 4;
  atomicAdd(ap + 0, v.x);
  atomicAdd(ap + 1, v.y);
  atomicAdd(ap + 2, v.z);
  atomicAdd(ap + 3, v.w);
}

// ---------------------------------------------------------------------------
// Stage 4: layer-1 fused GEMM with fp32 WMMA.
//   out = relu([agg1*inv_deg | h] (N x 256)  @  [Wl1; Wr1] (256 x 128) + b1)
// Block = 256 threads = 8 waves, computes a 32-node x 128-col tile.
// A-panel (32 x 256) staged in LDS (row stride 257 -> conflict-free banks).
// Each wave owns a 16-col slice and two row tiles; every B fragment load
// feeds TWO v_wmma_f32_16x16x4_f32 ops (2x reuse of global W traffic).
//
// Fragment layouts per ISA 7.12.2 (wave32):
//   A 16x4 f32:  lane -> m = lane&15, k = vgpr + 2*(lane>>4)
//   B  4x16 f32: lane -> n = lane&15, k = vgpr + 2*(lane>>4)
//   C/D 16x16:   lane -> n = lane&15, M = vgpr + 8*(lane>>4)
// ---------------------------------------------------------------------------
__global__ void __launch_bounds__(256) layer1_gemm_kernel(
    const float* __restrict__ agg1, const float* __restrict__ h,
    const float* __restrict__ deg,
    const float* __restrict__ Wl1, const float* __restrict__ Wr1,
    const float* __restrict__ b1, float* __restrict__ out, int N) {
  __shared__ float As[TILE_M][257];
  const int node_base = blockIdx.x * TILE_M;
  const int tid = threadIdx.x;

  // Cooperative load of the A panel: cols [0,128) = agg1 * inv_deg, [128,256) = h
  for (int i = tid; i < TILE_M * 256; i += 256) {
    int r = i >> 8;
    int c = i & 255;
    int node = node_base + r;
    float v = 0.0f;
    if (node < N) {
      if (c < DG) {
        float invd = 1.0f / fmaxf(deg[node], 1.0f);
        v = agg1[(size_t)node * DG + c] * invd;
      } else {
        v = h[(size_t)node * DG + (c - DG)];
      }
    }
    As[r][c] = v;
  }
  __syncthreads();

  const int wave = tid >> 5;
  const int lane = tid & 31;
  const int m = lane & 15;
  const int half = lane >> 4;
  const int ncol = wave * 16 + m;       // this lane's output column

  v8f acc0 = {};                        // rows node_base + [0,16)
  v8f acc1 = {};                        // rows node_base + [16,32)

  // K = [0,128): aggregated neighbors @ Wl1
#pragma unroll 4
  for (int k0 = 0; k0 < DG; k0 += 4) {
    const int kk = k0 + 2 * half;
    v2f b;
    b.x = Wl1[(size_t)(kk + 0) * DG + ncol];
    b.y = Wl1[(size_t)(kk + 1) * DG + ncol];
    v2f a0; a0.x = As[m][kk];      a0.y = As[m][kk + 1];
    v2f a1; a1.x = As[16 + m][kk]; a1.y = As[16 + m][kk + 1];
    acc0 = __builtin_amdgcn_wmma_f32_16x16x4_f32(false, a0, false, b, (short)0, acc0, false, false);
    acc1 = __builtin_amdgcn_wmma_f32_16x16x4_f32(false, a1, false, b, (short)0, acc1, false, false);
  }
  // K = [128,256): self features @ Wr1
#pragma unroll 4
  for (int k0 = 0; k0 < DG; k0 += 4) {
    const int kk = k0 + 2 * half;
    v2f b;
    b.x = Wr1[(size_t)(kk + 0) * DG + ncol];
    b.y = Wr1[(size_t)(kk + 1) * DG + ncol];
    v2f a0; a0.x = As[m][DG + kk];      a0.y = As[m][DG + kk + 1];
    v2f a1; a1.x = As[16 + m][DG + kk]; a1.y = As[16 + m][DG + kk + 1];
    acc0 = __builtin_amdgcn_wmma_f32_16x16x4_f32(false, a0, false, b, (short)0, acc0, false, false);
    acc1 = __builtin_amdgcn_wmma_f32_16x16x4_f32(false, a1, false, b, (short)0, acc1, false, false);
  }

  const float bias = b1[ncol];
  const int r0 = node_base + 8 * half;
  if (node_base + TILE_M <= N) {
    // Fast path: whole tile in range, unconditional coalesced stores.
#pragma unroll
    for (int v = 0; v < 8; ++v) {
      out[(size_t)(r0 + v) * DG + ncol]      = fmaxf(acc0[v] + bias, 0.0f);
      out[(size_t)(r0 + 16 + v) * DG + ncol] = fmaxf(acc1[v] + bias, 0.0f);
    }
  } else {
#pragma unroll
    for (int v = 0; v < 8; ++v) {
      int ra = r0 + v, rb = r0 + 16 + v;
      if (ra < N) out[(size_t)ra * DG + ncol] = fmaxf(acc0[v] + bias, 0.0f);
      if (rb < N) out[(size_t)rb * DG + ncol] = fmaxf(acc1[v] + bias, 0.0f);
    }
  }
}

// ---------------------------------------------------------------------------
// Host-side launch. Workspace layout (floats):
//   [0, N)        deg
//   [N, 4N)       agg0 (N x 3)
//   [4N, 132N)    agg1 (N x 128)
//   [132N, 260N)  h    (N x 128)
// ---------------------------------------------------------------------------
extern "C" void kernel_launch(void* const* d_in, const int* in_sizes, int n_in,
                              void* d_out, int out_size, void* d_ws, size_t ws_size,
                              hipStream_t stream) {
  const float* x    = (const float*)d_in[0];
  const int*   edge = (const int*)d_in[1];   // [2, E] int32
  const float* Wl0  = (const float*)d_in[2];
  const float* Wr0  = (const float*)d_in[3];
  const float* b0   = (const float*)d_in[4];
  const float* Wl1  = (const float*)d_in[5];
  const float* Wr1  = (const float*)d_in[6];
  const float* b1   = (const float*)d_in[7];
  const float* ln_g = (const float*)d_in[8];
  const float* ln_b = (const float*)d_in[9];

  const int N = in_sizes[0] / 3;
  const int E = in_sizes[1] / 2;
  const int* src = edge;
  const int* dst = edge + E;

  float* ws   = (float*)d_ws;
  float* deg  = ws;
  float* agg0 = ws + (size_t)N;
  float* agg1 = ws + (size_t)4 * N;
  float* h    = ws + (size_t)132 * N;
  float* out  = (float*)d_out;

  // Zero the accumulation buffers (deg, agg0, agg1); h/out are fully written.
  hipMemsetAsync(d_ws, 0, (size_t)132 * N * sizeof(float), stream);

  deg_agg0_kernel<<<(E + 255) / 256, 256, 0, stream>>>(src, dst, x, deg, agg0, E);

  layer0_node_kernel<<<N, DG, 0, stream>>>(x, agg0, deg, Wl0, Wr0, b0, ln_g, ln_b, h, N);

  long long ethreads = (long long)E * 32;
  layer1_agg_kernel<<<(int)((ethreads + 255) / 256), 256, 0, stream>>>(src, dst, h, agg1, E);

  layer1_gemm_kernel<<<(N + TILE_M - 1) / TILE_M, 256, 0, stream>>>(
      agg1, h, deg, Wl1, Wr1, b1, out, N);
}